// Model_73461120631179
// MI455X (gfx1250) — compile-verified
//
#include <hip/hip_runtime.h>
#include <math.h>

#define Bn 64
#define Nn 4096
#define Ln 128
#define En 64
#define Dn 128

typedef __attribute__((ext_vector_type(2))) float v2f;
typedef __attribute__((ext_vector_type(8))) float v8f;

static __device__ __forceinline__ float fxs()  { return 65536.0f; }
static __device__ __forceinline__ float ifxs() { return 1.0f/65536.0f; }
static __device__ __forceinline__ float fxg()  { return 262144.0f; }
static __device__ __forceinline__ float ifxg() { return 1.0f/262144.0f; }

// ---------------- Phase 0: weight prep (Wsum = We0+We1, WqT[c][d] = Wquat[d][c]) ----
__global__ __launch_bounds__(128)
void k_prep(const float* __restrict__ W_ev, const float* __restrict__ qr,
            const float* __restrict__ qi, const float* __restrict__ qj,
            const float* __restrict__ qk_, float* __restrict__ Wsum,
            float* __restrict__ WqT) {
  int i = blockIdx.x;      // 0..127
  int d2 = threadIdx.x;    // 0..127
  Wsum[i*Dn + d2] = W_ev[i*Dn + d2] + W_ev[(Dn+i)*Dn + d2];
  int c = i, d = d2;
  int br = d >> 5, bc = c >> 5, ii = d & 31, jj = c & 31;
  const float* Q[4] = {qr, qi, qj, qk_};
  const int   sel[4][4] = {{0,1,2,3},{1,0,3,2},{2,3,0,1},{3,2,1,0}};
  const float sgn[4][4] = {{1.f,-1.f,-1.f,-1.f},{1.f,1.f,-1.f,1.f},
                           {1.f,1.f,1.f,-1.f},{1.f,-1.f,1.f,1.f}};
  WqT[c*Dn + d] = sgn[br][bc] * Q[sel[br][bc]][ii*32 + jj];
}

// ---------------- Phase 3: temporal hyperedge features ------------------------------
__global__ __launch_bounds__(128)
void k_temp_he(const float* __restrict__ mark, const float* __restrict__ W_temp,
               const float* __restrict__ b_temp, float* __restrict__ temp_he) {
  int bl = blockIdx.x;        // b*L + l
  int d  = threadIdx.x;
  float mk = mark[bl];
  temp_he[(size_t)bl*Dn + d] = sinf(mk*W_temp[d] + b_temp[d]);
}

// ---------------- Phase 1: per-batch group stats (var_ctx, counts, q_aux, ctx.wm1) --
__global__ __launch_bounds__(256)
void k_group_stats(const float* __restrict__ xL, const float* __restrict__ msk,
                   const float* __restrict__ yM, const int* __restrict__ vidx,
                   const int* __restrict__ tidx, const float* __restrict__ W_obs,
                   const float* __restrict__ b_obs, const float* __restrict__ W_mem,
                   float* __restrict__ var_ctx, float* __restrict__ q_aux,
                   float* __restrict__ ctxdot) {
  __shared__ int sobs[En*Dn];
  __shared__ int hist[En*Ln];
  __shared__ int cnt[En];
  const int b = blockIdx.x;
  const int t = threadIdx.x;
  for (int i = t; i < En*Dn; i += 256) sobs[i] = 0;
  for (int i = t; i < En*Ln; i += 256) hist[i] = 0;
  if (t < En) cnt[t] = 0;
  __syncthreads();
  const int lane = t & 31, wave = t >> 5;
  float w0[4], w1[4], bb[4];
  for (int c = 0; c < 4; ++c) {
    int d = c*32 + lane;
    w0[c] = W_obs[d]; w1[c] = W_obs[Dn + d]; bb[c] = b_obs[d];
  }
  for (int n = wave; n < Nn; n += 8) {
    float m = msk[(size_t)b*Nn + n];
    if (m == 0.0f) continue;
    float x = xL[(size_t)b*Nn + n];
    float xin2 = 1.0f - m + yM[(size_t)b*Nn + n];
    int e = vidx[(size_t)b*Nn + n];
    int tt = tidx[(size_t)b*Nn + n];
    for (int c = 0; c < 4; ++c) {
      float o = fmaxf(x*w0[c] + xin2*w1[c] + bb[c], 0.0f);
      atomicAdd(&sobs[e*Dn + c*32 + lane], __float2int_rn(o*fxs()));
    }
    if (lane == 0) { atomicAdd(&hist[e*Ln + tt], 1); atomicAdd(&cnt[e], 1); }
  }
  __syncthreads();
  for (int i = t; i < En*Dn; i += 256) {
    int e = i / Dn;
    int cc = cnt[e] > 1 ? cnt[e] : 1;
    var_ctx[(size_t)b*En*Dn + i] = (float)sobs[i] * ifxs() / (float)cc;
  }
  for (int i = t; i < En*Ln; i += 256)
    q_aux[(size_t)b*En*Ln + i] = (float)hist[i];
  __syncthreads();
  if (t < En) {
    int cc = cnt[t] > 1 ? cnt[t] : 1;
    float acc = 0.0f;
    for (int d = 0; d < Dn; ++d) acc += (float)sobs[t*Dn + d] * W_mem[Dn + d];
    ctxdot[b*En + t] = acc * ifxs() / (float)cc;
  }
}

// ---------------- Phase 2: gated group sums S1, S2, g; A2 = -g*var_ctx --------------
__global__ __launch_bounds__(256)
void k_gates(const float* __restrict__ xL, const float* __restrict__ msk,
             const float* __restrict__ yM, const int* __restrict__ vidx,
             const float* __restrict__ W_obs, const float* __restrict__ b_obs,
             const float* __restrict__ W_mem, const float* __restrict__ b_mem,
             const float* __restrict__ rls_p, const float* __restrict__ els_p,
             const float* __restrict__ var_ctx, const float* __restrict__ ctxdot,
             float* __restrict__ S1, float* __restrict__ S2,
             float* __restrict__ gbuf, float* __restrict__ A2) {
  __shared__ int s1[En*Dn];
  __shared__ int s2[En*Dn];
  __shared__ int gg[En];
  __shared__ float ctxd[En];
  const int b = blockIdx.x, t = threadIdx.x;
  for (int i = t; i < En*Dn; i += 256) { s1[i] = 0; s2[i] = 0; }
  if (t < En) { gg[t] = 0; ctxd[t] = ctxdot[b*En + t]; }
  __syncthreads();
  const int lane = t & 31, wave = t >> 5;
  float w0[4], w1[4], bb[4], ws[4];
  for (int c = 0; c < 4; ++c) {
    int d = c*32 + lane;
    w0[c] = W_obs[d]; w1[c] = W_obs[Dn + d]; bb[c] = b_obs[d];
    ws[c] = W_mem[d] + W_mem[Dn + d];
  }
  const float bm = b_mem[0];
  const float rk = expf(rls_p[0]) - 1.0f;   // retain coefficient
  for (int n = wave; n < Nn; n += 8) {
    float m = msk[(size_t)b*Nn + n];
    if (m == 0.0f) continue;
    float x = xL[(size_t)b*Nn + n];
    float xin2 = 1.0f - m + yM[(size_t)b*Nn + n];
    int e = vidx[(size_t)b*Nn + n];
    float o[4], p = 0.0f;
    for (int c = 0; c < 4; ++c) {
      o[c] = fmaxf(x*w0[c] + xin2*w1[c] + bb[c], 0.0f);
      p += o[c]*ws[c];
    }
    for (int off = 16; off > 0; off >>= 1) p += __shfl_xor(p, off, 32);
    float mem = p - ctxd[e] + bm;
    float sg  = 1.0f / (1.0f + expf(-mem));        // sigmoid(membrane)
    float rg  = 1.0f - rk*(1.0f - sg);             // retain gate
    for (int c = 0; c < 4; ++c) {
      atomicAdd(&s1[e*Dn + c*32 + lane], __float2int_rn(rg*o[c]*fxs()));
      atomicAdd(&s2[e*Dn + c*32 + lane], __float2int_rn(sg*o[c]*fxs()));
    }
    if (lane == 0) atomicAdd(&gg[e], __float2int_rn(sg*fxg()));
  }
  __syncthreads();
  const float ek = expf(els_p[0]);                 // event scale (pulled out of sums)
  for (int i = t; i < En*Dn; i += 256) {
    int e = i / Dn;
    float gv = ek * (float)gg[e] * ifxg();
    S1[(size_t)b*En*Dn + i] = (float)s1[i] * ifxs();
    S2[(size_t)b*En*Dn + i] = ek * (float)s2[i] * ifxs();
    A2[(size_t)b*En*Dn + i] = -gv * var_ctx[(size_t)b*En*Dn + i];
  }
  if (t < En) gbuf[b*En + t] = ek * (float)gg[t] * ifxg();
}

// ---------------- WMMA fp32 16x16x4 tile helper -------------------------------------
// One wave computes one 16x16 fp32 tile: acc += A(16xK) * B(KxN-tile)
// A layout (ISA): lanes 0-15 row M, v2f = {K=2*half, K=2*half+1}; B mirrored per column.
template<bool BT>
__device__ __forceinline__ v8f wmma_acc(v8f acc, const float* __restrict__ Am, int lda,
                                        const float* __restrict__ Bm, int ldb, int K) {
  const int lane = threadIdx.x & 31;
  const int half = lane >> 4;
  const int lr   = lane & 15;
  for (int k0 = 0; k0 < K; k0 += 4) {
    int ka = k0 + half*2;
    v2f a, bv;
    a.x = Am[lr*lda + ka];
    a.y = Am[lr*lda + ka + 1];
    if (BT) { bv.x = Bm[lr*ldb + ka];  bv.y = Bm[lr*ldb + ka + 1]; }
    else    { bv.x = Bm[ka*ldb + lr];  bv.y = Bm[(ka+1)*ldb + lr]; }
    acc = __builtin_amdgcn_wmma_f32_16x16x4_f32(false, a, false, bv, (short)0, acc,
                                                false, false);
  }
  return acc;
}

// ---------------- Phase 4: var_feat assembly (three fused WMMA GEMMs) ----------------
__global__ __launch_bounds__(32)
void k_varfeat(const float* __restrict__ S2, const float* __restrict__ A2,
               const float* __restrict__ q_aux, const float* __restrict__ Wsum,
               const float* __restrict__ W_ev, const float* __restrict__ temp_he,
               const float* __restrict__ var_w, const float* __restrict__ b_ev,
               const float* __restrict__ S1, const float* __restrict__ gbuf,
               float* __restrict__ var_feat) {
  int b = blockIdx.x, tm = blockIdx.y, tn = blockIdx.z;
  v8f acc = {};
  acc = wmma_acc<false>(acc, S2 + (size_t)b*En*Dn + tm*16*Dn, Dn, Wsum + tn*16, Dn, Dn);
  acc = wmma_acc<false>(acc, A2 + (size_t)b*En*Dn + tm*16*Dn, Dn,
                        W_ev + Dn*Dn + tn*16, Dn, Dn);
  acc = wmma_acc<false>(acc, q_aux + (size_t)b*En*Ln + tm*16*Ln, Ln,
                        temp_he + (size_t)b*Ln*Dn + tn*16, Dn, Ln);
  const int lane = threadIdx.x & 31, half = lane >> 4, lr = lane & 15;
  const float invL = 1.0f / (float)Ln;
  for (int v = 0; v < 8; ++v) {
    int e = tm*16 + half*8 + v;
    int d = tn*16 + lr;
    float val = fmaxf(var_w[e*Dn + d], 0.0f)
              + invL * (S1[(size_t)b*En*Dn + e*Dn + d] + gbuf[b*En + e]*b_ev[d] + acc[v]);
    var_feat[(size_t)b*En*Dn + e*Dn + d] = val;
  }
}

// ---------------- Phase 5: q/k/v projections -----------------------------------------
__global__ __launch_bounds__(32)
void k_qkv(const float* __restrict__ var_feat,
           const float* __restrict__ Wq, const float* __restrict__ bq,
           const float* __restrict__ Wk, const float* __restrict__ bk,
           const float* __restrict__ Wv, const float* __restrict__ bv,
           float* __restrict__ qkv) {
  int b = blockIdx.x;
  int proj = blockIdx.y >> 2, tm = blockIdx.y & 3, tn = blockIdx.z;
  const float* W    = proj == 0 ? Wq : (proj == 1 ? Wk : Wv);
  const float* bias = proj == 0 ? bq : (proj == 1 ? bk : bv);
  v8f acc = {};
  acc = wmma_acc<false>(acc, var_feat + (size_t)b*En*Dn + tm*16*Dn, Dn, W + tn*16, Dn, Dn);
  const int lane = threadIdx.x & 31, half = lane >> 4, lr = lane & 15;
  for (int v = 0; v < 8; ++v) {
    int e = tm*16 + half*8 + v, d = tn*16 + lr;
    qkv[((size_t)proj*Bn + b)*En*Dn + e*Dn + d] = acc[v] + bias[d];
  }
}

// ---------------- Phase 6: scores = q k^T merged with q_aux q_aux^T ------------------
__global__ __launch_bounds__(32)
void k_scores(const float* __restrict__ qkv, const float* __restrict__ q_aux,
              const float* __restrict__ thr_p, const float* __restrict__ mc_p,
              float* __restrict__ scores) {
  int b = blockIdx.x, tm = blockIdx.y, tn = blockIdx.z;
  const float* qb = qkv + (size_t)b*En*Dn;
  const float* kb = qkv + ((size_t)Bn + b)*En*Dn;
  v8f a1 = {}, a2 = {};
  a1 = wmma_acc<true>(a1, qb + tm*16*Dn, Dn, kb + tn*16*Dn, Dn, Dn);
  a2 = wmma_acc<true>(a2, q_aux + (size_t)b*En*Ln + tm*16*Ln, Ln,
                      q_aux + (size_t)b*En*Ln + tn*16*Ln, Ln, Ln);
  const float sD = rsqrtf((float)Dn), sL = rsqrtf((float)Ln);
  const float thr = thr_p[0], mc = mc_p[0];
  const int lane = threadIdx.x & 31, half = lane >> 4, lr = lane & 15;
  for (int v = 0; v < 8; ++v) {
    int e = tm*16 + half*8 + v, f = tn*16 + lr;
    float s  = a1[v] * sD;
    float sa = a2[v] * sL;
    if (sa != 0.0f && s > thr) s = (1.0f - mc)*s + mc*sa;
    scores[(size_t)b*En*En + e*En + f] = s;
  }
}

// ---------------- Phase 7: row softmax over E=64 -------------------------------------
__global__ __launch_bounds__(64)
void k_softmax(float* __restrict__ scores) {
  __shared__ float red[64];
  int row = blockIdx.x, t = threadIdx.x;
  float s = scores[(size_t)row*En + t];
  red[t] = s; __syncthreads();
  for (int off = 32; off > 0; off >>= 1) {
    if (t < off) red[t] = fmaxf(red[t], red[t + off]);
    __syncthreads();
  }
  float mx = red[0]; __syncthreads();
  float ex = expf(s - mx);
  red[t] = ex; __syncthreads();
  for (int off = 32; off > 0; off >>= 1) {
    if (t < off) red[t] += red[t + off];
    __syncthreads();
  }
  scores[(size_t)row*En + t] = ex / red[0];
}

// ---------------- Phase 8: attn_out = attn @ v ---------------------------------------
__global__ __launch_bounds__(32)
void k_attnout(const float* __restrict__ scores, const float* __restrict__ qkv,
               float* __restrict__ attn_out) {
  int b = blockIdx.x, tm = blockIdx.y, tn = blockIdx.z;
  const float* vb = qkv + ((size_t)2*Bn + b)*En*Dn;
  v8f acc = {};
  acc = wmma_acc<false>(acc, scores + (size_t)b*En*En + tm*16*En, En, vb + tn*16, Dn, En);
  const int lane = threadIdx.x & 31, half = lane >> 4, lr = lane & 15;
  for (int v = 0; v < 8; ++v)
    attn_out[(size_t)b*En*Dn + (tm*16 + half*8 + v)*Dn + tn*16 + lr] = acc[v];
}

// ---------------- Phase 9: out = attn_out @ Wquat^T + q_bias -------------------------
__global__ __launch_bounds__(32)
void k_final(const float* __restrict__ attn_out, const float* __restrict__ WqT,
             const float* __restrict__ q_bias, float* __restrict__ out) {
  int b = blockIdx.x, tm = blockIdx.y, tn = blockIdx.z;
  v8f acc = {};
  acc = wmma_acc<false>(acc, attn_out + (size_t)b*En*Dn + tm*16*Dn, Dn, WqT + tn*16, Dn, Dn);
  const int lane = threadIdx.x & 31, half = lane >> 4, lr = lane & 15;
  for (int v = 0; v < 8; ++v) {
    int e = tm*16 + half*8 + v, d = tn*16 + lr;
    out[(size_t)b*En*Dn + e*Dn + d] = acc[v] + q_bias[d];
  }
}

extern "C" void kernel_launch(void* const* d_in, const int* in_sizes, int n_in,
                              void* d_out, int out_size, void* d_ws, size_t ws_size,
                              hipStream_t stream) {
  const float* xL    = (const float*)d_in[0];
  const float* msk   = (const float*)d_in[1];
  const float* yM    = (const float*)d_in[2];
  const float* mark  = (const float*)d_in[3];
  const float* W_obs = (const float*)d_in[4];
  const float* b_obs = (const float*)d_in[5];
  const float* W_tmp = (const float*)d_in[6];
  const float* b_tmp = (const float*)d_in[7];
  const float* var_w = (const float*)d_in[8];
  const float* W_mem = (const float*)d_in[9];
  const float* b_mem = (const float*)d_in[10];
  const float* W_ev  = (const float*)d_in[11];
  const float* b_ev  = (const float*)d_in[12];
  const float* rls   = (const float*)d_in[13];
  const float* els   = (const float*)d_in[14];
  const float* Wq    = (const float*)d_in[15];
  const float* bq    = (const float*)d_in[16];
  const float* Wk    = (const float*)d_in[17];
  const float* bk    = (const float*)d_in[18];
  const float* Wv    = (const float*)d_in[19];
  const float* bv    = (const float*)d_in[20];
  const float* thr   = (const float*)d_in[21];
  const float* mc    = (const float*)d_in[22];
  const float* qr    = (const float*)d_in[23];
  const float* qi    = (const float*)d_in[24];
  const float* qj    = (const float*)d_in[25];
  const float* qk_   = (const float*)d_in[26];
  const float* qbias = (const float*)d_in[27];
  const int*   vidx  = (const int*)d_in[28];
  const int*   tidx  = (const int*)d_in[29];
  float* out = (float*)d_out;

  float* w = (float*)d_ws;
  float* var_ctx  = w; w += (size_t)Bn*En*Dn;
  float* q_aux    = w; w += (size_t)Bn*En*Ln;
  float* ctxdot   = w; w += (size_t)Bn*En;
  float* S1       = w; w += (size_t)Bn*En*Dn;
  float* S2       = w; w += (size_t)Bn*En*Dn;
  float* gbuf     = w; w += (size_t)Bn*En;
  float* A2       = w; w += (size_t)Bn*En*Dn;
  float* temp_he  = w; w += (size_t)Bn*Ln*Dn;
  float* Wsum     = w; w += (size_t)Dn*Dn;
  float* WqT      = w; w += (size_t)Dn*Dn;
  float* var_feat = w; w += (size_t)Bn*En*Dn;
  float* qkv      = w; w += (size_t)3*Bn*En*Dn;
  float* scores   = w; w += (size_t)Bn*En*En;
  float* attn_out = w; w += (size_t)Bn*En*Dn;

  k_prep<<<dim3(Dn), dim3(Dn), 0, stream>>>(W_ev, qr, qi, qj, qk_, Wsum, WqT);
  k_temp_he<<<dim3(Bn*Ln), dim3(Dn), 0, stream>>>(mark, W_tmp, b_tmp, temp_he);
  k_group_stats<<<dim3(Bn), dim3(256), 0, stream>>>(xL, msk, yM, vidx, tidx, W_obs,
                                                    b_obs, W_mem, var_ctx, q_aux, ctxdot);
  k_gates<<<dim3(Bn), dim3(256), 0, stream>>>(xL, msk, yM, vidx, W_obs, b_obs, W_mem,
                                              b_mem, rls, els, var_ctx, ctxdot,
                                              S1, S2, gbuf, A2);
  k_varfeat<<<dim3(Bn, 4, 8), dim3(32), 0, stream>>>(S2, A2, q_aux, Wsum, W_ev, temp_he,
                                                     var_w, b_ev, S1, gbuf, var_feat);
  k_qkv<<<dim3(Bn, 12, 8), dim3(32), 0, stream>>>(var_feat, Wq, bq, Wk, bk, Wv, bv, qkv);
  k_scores<<<dim3(Bn, 4, 4), dim3(32), 0, stream>>>(qkv, q_aux, thr, mc, scores);
  k_softmax<<<dim3(Bn*En), dim3(64), 0, stream>>>(scores);
  k_attnout<<<dim3(Bn, 4, 8), dim3(32), 0, stream>>>(scores, qkv, attn_out);
  k_final<<<dim3(Bn, 4, 8), dim3(32), 0, stream>>>(attn_out, WqT, qbias, out);
}